// GCN_11682311045288
// MI455X (gfx1250) — compile-verified
//
#include <hip/hip_runtime.h>
#include <hip/hip_bf16.h>

// GCN 2-layer: z = GCNConv(relu(GCNConv(x,W1,b1)), W2, b2) with symmetric
// deg^-1/2 normalization and self-loops.
// Strategy: f32 WMMA (v_wmma_f32_16x16x4_f32) for the dense XW GEMMs (exact
// fp32 semantics, GEMMs are tiny vs. the 2 GB edge traffic), precomputed
// per-edge norm (shared across both layers), float4 gathers + f32 global
// atomics for the scatter (the bandwidth-bound part, ~85us at 23.3 TB/s).

typedef __attribute__((ext_vector_type(2))) float v2f;
typedef __attribute__((ext_vector_type(8))) float v8f;

#define TPB 256

// ---------------- degree / normalization ----------------

__global__ void k_fill1(float* __restrict__ p, long long n) {
  long long i = blockIdx.x * (long long)TPB + threadIdx.x;
  if (i < n) p[i] = 1.0f;  // self-loop contributes 1 to in-degree
}

__global__ void k_deg(const long long* __restrict__ dst, float* __restrict__ deg,
                      long long E) {
  long long e = blockIdx.x * (long long)TPB + threadIdx.x;
  if (e < E) atomicAdd(&deg[dst[e]], 1.0f);
}

__global__ void k_dinv(const float* __restrict__ deg, float* __restrict__ dinv,
                       long long n) {
  long long i = blockIdx.x * (long long)TPB + threadIdx.x;
  if (i < n) dinv[i] = 1.0f / sqrtf(deg[i]);  // deg >= 1 always (self-loops)
}

__global__ void k_norm(const long long* __restrict__ src,
                       const long long* __restrict__ dst,
                       const float* __restrict__ dinv,
                       float* __restrict__ norm, long long E) {
  long long e = blockIdx.x * (long long)TPB + threadIdx.x;
  if (e < E) norm[e] = dinv[src[e]] * dinv[dst[e]];
}

// ---------------- dense GEMM: C[M x 64] = A[M x K] * B[K x 64] -------------
// Wave32 WMMA f32 16x16x4. 8 waves/block: 2 row-tiles x 4 col-tiles = 32 rows
// per block. Requires M % 32 == 0 (100000 = 3125*32).
// A frag (16x4 f32, 2 VGPR): v0 = K=k0 (lanes 0-15) / k0+2 (lanes 16-31),
//                            v1 = K=k0+1 / k0+3.
// B frag (4x16 f32, 2 VGPR): v0 = rows k0 / k0+2 striped over lane halves,
//                            v1 = rows k0+1 / k0+3 (mirrors the C layout rule).
template <int K>
__global__ void k_gemm64(const float* __restrict__ A, const float* __restrict__ B,
                         float* __restrict__ C) {
  const int lane = threadIdx.x & 31;
  const int wave = threadIdx.x >> 5;
  const int rowTile = blockIdx.x * 32 + (wave >> 2) * 16;
  const int colTile = (wave & 3) * 16;
  const int half = lane >> 4;   // 0: lanes 0-15, 1: lanes 16-31
  const int l15 = lane & 15;
  const int arow = rowTile + l15;

  v8f acc = {};
#pragma unroll
  for (int k0 = 0; k0 < K; k0 += 4) {
    const int ka = k0 + half * 2;
    v2f a;
    a.x = A[arow * K + ka];
    a.y = A[arow * K + ka + 1];
    v2f b;
    b.x = B[ka * 64 + colTile + l15];
    b.y = B[(ka + 1) * 64 + colTile + l15];
    acc = __builtin_amdgcn_wmma_f32_16x16x4_f32(
        /*neg_a=*/false, a, /*neg_b=*/false, b,
        /*c_mod=*/(short)0, acc, /*reuse_a=*/false, /*reuse_b=*/false);
  }

#pragma unroll
  for (int r = 0; r < 8; ++r) {
    const int orow = rowTile + r + half * 8;
    C[orow * 64 + colTile + l15] = acc[r];
  }
}

// ---------------- aggregation ----------------

// out[i][f] = bias[f] + dinv[i]^2 * feat[i][f]   (bias + self-loop edge)
__global__ void k_self(const float* __restrict__ feat, const float* __restrict__ bias,
                       const float* __restrict__ dinv, float* __restrict__ out,
                       long long total /* = N*64 */) {
  long long g = blockIdx.x * (long long)TPB + threadIdx.x;
  if (g >= total) return;
  const long long i = g >> 6;
  const int f = (int)(g & 63);
  const float di = dinv[i];
  out[g] = bias[f] + di * di * feat[g];
}

// scatter-add: out[dst[e]][:] += norm[e] * feat[src[e]][:]
// 16 lanes per edge, 4 features per lane (float4 gather, 4x f32 atomics).
__global__ void k_scatter(const float* __restrict__ feat,
                          const float* __restrict__ norm,
                          const long long* __restrict__ src,
                          const long long* __restrict__ dst,
                          float* __restrict__ out, long long E) {
  long long g = blockIdx.x * (long long)TPB + threadIdx.x;
  if (g >= E * 16) return;
  const long long e = g >> 4;
  const int c = (int)(g & 15);
  const long long s = src[e];
  const long long d = dst[e];
  const float w = norm[e];
  const float4 v = *(const float4*)&feat[s * 64 + c * 4];
  float* o = &out[d * 64 + c * 4];
  atomicAdd(o + 0, v.x * w);
  atomicAdd(o + 1, v.y * w);
  atomicAdd(o + 2, v.z * w);
  atomicAdd(o + 3, v.w * w);
}

__global__ void k_relu(float* __restrict__ p, long long n) {
  long long i = blockIdx.x * (long long)TPB + threadIdx.x;
  if (i < n) p[i] = fmaxf(p[i], 0.0f);
}

// ---------------- launch ----------------

static inline long long cdiv(long long a, long long b) { return (a + b - 1) / b; }

extern "C" void kernel_launch(void* const* d_in, const int* in_sizes, int n_in,
                              void* d_out, int out_size, void* d_ws, size_t ws_size,
                              hipStream_t stream) {
  const float* x = (const float*)d_in[0];                 // [N,128]
  const long long* ei = (const long long*)d_in[1];        // [2,E] int64
  const float* W1 = (const float*)d_in[2];                // [128,64]
  const float* b1 = (const float*)d_in[3];                // [64]
  const float* W2 = (const float*)d_in[4];                // [64,64]
  const float* b2 = (const float*)d_in[5];                // [64]
  float* out = (float*)d_out;                             // [N,64]

  const long long N = (long long)in_sizes[0] / 128;
  const long long E = (long long)in_sizes[1] / 2;
  const long long* src = ei;
  const long long* dsti = ei + E;

  // workspace carve-up (all chunks are 16B-aligned: N*4 and E*4 are)
  char* ws = (char*)d_ws;
  float* deg  = (float*)ws;                 ws += N * 4;
  float* dinv = (float*)ws;                 ws += N * 4;
  float* norm = (float*)ws;                 ws += E * 4;
  float* xw   = (float*)ws;                 ws += N * 64 * 4;   // also reused as hw
  float* agg1 = (float*)ws;

  // normalization (computed once, reused by both layers)
  k_fill1<<<cdiv(N, TPB), TPB, 0, stream>>>(deg, N);
  k_deg<<<cdiv(E, TPB), TPB, 0, stream>>>(dsti, deg, E);
  k_dinv<<<cdiv(N, TPB), TPB, 0, stream>>>(deg, dinv, N);
  k_norm<<<cdiv(E, TPB), TPB, 0, stream>>>(src, dsti, dinv, norm, E);

  // layer 1: xw = x @ W1 ; agg1 = b1 + dinv^2*xw + scatter ; relu
  k_gemm64<128><<<N / 32, TPB, 0, stream>>>(x, W1, xw);
  k_self<<<cdiv(N * 64, TPB), TPB, 0, stream>>>(xw, b1, dinv, agg1, N * 64);
  k_scatter<<<cdiv(E * 16, TPB), TPB, 0, stream>>>(xw, norm, src, dsti, agg1, E);
  k_relu<<<cdiv(N * 64, TPB), TPB, 0, stream>>>(agg1, N * 64);

  // layer 2: hw = agg1 @ W2 (reuse xw buffer) ; out = b2 + dinv^2*hw + scatter
  k_gemm64<64><<<N / 32, TPB, 0, stream>>>(agg1, W2, xw);
  k_self<<<cdiv(N * 64, TPB), TPB, 0, stream>>>(xw, b2, dinv, out, N * 64);
  k_scatter<<<cdiv(E * 16, TPB), TPB, 0, stream>>>(xw, norm, src, dsti, out, E);
}